// GATModel_7060926235314
// MI455X (gfx1250) — compile-verified
//
#include <hip/hip_runtime.h>

typedef float v2f __attribute__((ext_vector_type(2)));
typedef float v8f __attribute__((ext_vector_type(8)));

#define NEG_SLOPE 0.2f

// ---- monotone float<->uint flip for atomicMax on floats -------------------
__device__ __forceinline__ unsigned fflip(float f) {
  unsigned u = __float_as_uint(f);
  return (u & 0x80000000u) ? ~u : (u | 0x80000000u);
}
__device__ __forceinline__ float funflip(unsigned u) {
  return __uint_as_float((u & 0x80000000u) ? (u & 0x7FFFFFFFu) : ~u);
}

// ---- generic 32-bit fill ---------------------------------------------------
__global__ void init_u32(unsigned* __restrict__ p, unsigned v, long long n) {
  long long i = (long long)blockIdx.x * blockDim.x + threadIdx.x;
  if (i < n) p[i] = v;
}

// ---- f32 WMMA GEMM: C[M,N] = A[M,K] @ B[K,N], row-major --------------------
// One wave computes a 16-row x (16*NT)-col strip of C.
template <int K, int N, int NT>
__global__ void gemm_wmma(const float* __restrict__ A, const float* __restrict__ B,
                          float* __restrict__ C, int M) {
  const int waves_per_block = blockDim.x >> 5;
  int wid = blockIdx.x * waves_per_block + (threadIdx.x >> 5);
  int lane = threadIdx.x & 31;
  const int col_groups = N / (16 * NT);
  int total = (M / 16) * col_groups;
  if (wid >= total) return;
  int mt = wid / col_groups;
  int ng = wid % col_groups;
  int row = mt * 16 + (lane & 15);
  int colbase = ng * (16 * NT) + (lane & 15);
  int ksel = (lane >> 4) << 1;  // lanes 0-15: K=kk,kk+1 ; lanes 16-31: K=kk+2,kk+3

  v8f zero = {};
  v8f acc[NT];
#pragma unroll
  for (int t = 0; t < NT; ++t) acc[t] = zero;

  const float* Arow = A + (size_t)row * K;
  for (int kk = 0; kk < K; kk += 4) {
    v2f a;
    a.x = Arow[kk + ksel];
    a.y = Arow[kk + ksel + 1];
#pragma unroll
    for (int t = 0; t < NT; ++t) {
      v2f b;
      b.x = B[(size_t)(kk + ksel) * N + colbase + t * 16];
      b.y = B[(size_t)(kk + ksel + 1) * N + colbase + t * 16];
      acc[t] = __builtin_amdgcn_wmma_f32_16x16x4_f32(
          false, a, false, b, (short)0, acc[t], false, false);
    }
  }
  // C/D layout: VGPR r -> row r (lanes 0-15) / row r+8 (lanes 16-31)
  int rbase = (lane >> 4) << 3;
#pragma unroll
  for (int t = 0; t < NT; ++t)
#pragma unroll
    for (int r = 0; r < 8; ++r)
      C[(size_t)(mt * 16 + rbase + r) * N + colbase + t * 16] = acc[t][r];
}

// ---- per-node attention logits: als/ald[n,h] = dot(h[n, h*64:(h+1)*64], a) -
template <int C, int H>  // C = H*64
__global__ void attn_logits(const float* __restrict__ h, const float* __restrict__ a_src,
                            const float* __restrict__ a_dst, float* __restrict__ als,
                            float* __restrict__ ald, int N) {
  int wid = blockIdx.x * (blockDim.x >> 5) + (threadIdx.x >> 5);  // one wave per node
  int lane = threadIdx.x & 31;
  if (wid >= N) return;
  const int PER = C / 32;        // floats per lane (8 for C=256, 2 for C=64)
  const int LPH = 64 / PER;      // lanes covering one head
  const float* row = h + (size_t)wid * C;
  int base = lane * PER;
  float ps = 0.f, pd = 0.f;
#pragma unroll
  for (int i = 0; i < PER; ++i) {
    float v = row[base + i];
    ps += v * a_src[base + i];
    pd += v * a_dst[base + i];
  }
#pragma unroll
  for (int off = 1; off < LPH; off <<= 1) {
    ps += __shfl_xor(ps, off, 32);
    pd += __shfl_xor(pd, off, 32);
  }
  int hd = lane / LPH;
  if ((lane % LPH) == 0) {
    als[wid * H + hd] = ps;
    ald[wid * H + hd] = pd;
  }
}

// ---- edge pass 1: segment max (atomicMax on flipped uint) ------------------
template <int H>
__global__ void edge_max(const int* __restrict__ esrc, const int* __restrict__ edst,
                         int E, int N, const float* __restrict__ als,
                         const float* __restrict__ ald, unsigned* __restrict__ m_u) {
  long long ETH = (long long)(E + N) * H;
  long long t = (long long)blockIdx.x * blockDim.x + threadIdx.x;
  if (t >= ETH) return;
  int e = (int)(t / H), hd = (int)(t % H);
  int s, d;
  if (e < E) { s = esrc[e]; d = edst[e]; } else { s = d = e - E; }
  float v = als[s * H + hd] + ald[d * H + hd];
  v = v > 0.f ? v : NEG_SLOPE * v;
  atomicMax(&m_u[d * H + hd], fflip(v));
}

// ---- edge pass 2: segment sum of exp(e - m) --------------------------------
template <int H>
__global__ void edge_sum(const int* __restrict__ esrc, const int* __restrict__ edst,
                         int E, int N, const float* __restrict__ als,
                         const float* __restrict__ ald, const unsigned* __restrict__ m_u,
                         float* __restrict__ z) {
  long long ETH = (long long)(E + N) * H;
  long long t = (long long)blockIdx.x * blockDim.x + threadIdx.x;
  if (t >= ETH) return;
  int e = (int)(t / H), hd = (int)(t % H);
  int s, d;
  if (e < E) { s = esrc[e]; d = edst[e]; } else { s = d = e - E; }
  float v = als[s * H + hd] + ald[d * H + hd];
  v = v > 0.f ? v : NEG_SLOPE * v;
  float x = __expf(v - funflip(m_u[d * H + hd]));
  atomicAdd(&z[d * H + hd], x);
}

// ---- edge pass 3: out[dst] += alpha * hlin[src] (L2-resident scatter) ------
template <int C, int H>
__global__ void edge_scatter(const int* __restrict__ esrc, const int* __restrict__ edst,
                             int E, int N, const float* __restrict__ als,
                             const float* __restrict__ ald, const unsigned* __restrict__ m_u,
                             const float* __restrict__ z, const float* __restrict__ hlin,
                             float* __restrict__ out) {
  const int EPB = 256 / C;  // edges handled per block
  int f = threadIdx.x % C;
  int eo = threadIdx.x / C;
  int hd = f / (C / H);
  long long ET = (long long)E + N;
  long long e0 = (long long)blockIdx.x * EPB + eo;
  if (e0 >= ET) return;
  int e = (int)e0;
  int s, d;
  if (e < E) { s = esrc[e]; d = edst[e]; } else { s = d = e - E; }
  float v = als[s * H + hd] + ald[d * H + hd];
  v = v > 0.f ? v : NEG_SLOPE * v;
  float alpha = __expf(v - funflip(m_u[d * H + hd])) / (z[d * H + hd] + 1e-16f);
  atomicAdd(&out[(size_t)d * C + f], alpha * hlin[(size_t)s * C + f]);
}

// ---- bias + relu (layers 1 & 2) -------------------------------------------
__global__ void bias_relu(float* __restrict__ buf, const float* __restrict__ b,
                          int C, long long total) {
  long long i = (long long)blockIdx.x * blockDim.x + threadIdx.x;
  if (i >= total) return;
  float v = buf[i] + b[(int)(i % C)];
  buf[i] = v > 0.f ? v : 0.f;
}

// ---- final: emb = agg + b3 ; preds = log_softmax(emb) ; one wave per node --
__global__ void finalize(float* __restrict__ emb, const float* __restrict__ b3,
                         float* __restrict__ preds, int N) {
  int wid = blockIdx.x * (blockDim.x >> 5) + (threadIdx.x >> 5);
  int lane = threadIdx.x & 31;
  if (wid >= N) return;
  size_t base = (size_t)wid * 64;
  float v0 = emb[base + lane] + b3[lane];
  float v1 = emb[base + lane + 32] + b3[lane + 32];
  float mx = fmaxf(v0, v1);
#pragma unroll
  for (int off = 16; off >= 1; off >>= 1) mx = fmaxf(mx, __shfl_xor(mx, off, 32));
  float s = __expf(v0 - mx) + __expf(v1 - mx);
#pragma unroll
  for (int off = 16; off >= 1; off >>= 1) s += __shfl_xor(s, off, 32);
  float l = __logf(s) + mx;
  emb[base + lane] = v0;
  emb[base + lane + 32] = v1;
  preds[base + lane] = v0 - l;
  preds[base + lane + 32] = v1 - l;
}

static inline int cdiv(long long a, long long b) { return (int)((a + b - 1) / b); }

extern "C" void kernel_launch(void* const* d_in, const int* in_sizes, int n_in,
                              void* d_out, int out_size, void* d_ws, size_t ws_size,
                              hipStream_t stream) {
  const float* x   = (const float*)d_in[0];
  const int*   ei  = (const int*)d_in[1];
  const float* W1  = (const float*)d_in[2];
  const float* a1s = (const float*)d_in[3];
  const float* a1d = (const float*)d_in[4];
  const float* b1  = (const float*)d_in[5];
  const float* W2  = (const float*)d_in[6];
  const float* a2s = (const float*)d_in[7];
  const float* a2d = (const float*)d_in[8];
  const float* b2  = (const float*)d_in[9];
  const float* W3  = (const float*)d_in[10];
  const float* a3s = (const float*)d_in[11];
  const float* a3d = (const float*)d_in[12];
  const float* b3  = (const float*)d_in[13];

  const int Nn = in_sizes[0] / 128;   // 80000
  const int E  = in_sizes[1] / 2;     // 1280000
  const int ET = E + Nn;              // with self-loops
  const int* esrc = ei;
  const int* edst = ei + E;

  // workspace layout (L2-residency plan: bufA + bufB = 164 MB <= 192 MB L2)
  float* p = (float*)d_ws;
  float*    bufA = p;             p += (size_t)Nn * 256;  // hlin of current layer
  float*    bufB = p;             p += (size_t)Nn * 256;  // aggregate / next input
  float*    als  = p;             p += (size_t)Nn * 4;
  float*    ald  = p;             p += (size_t)Nn * 4;
  unsigned* m_u  = (unsigned*)p;  p += (size_t)Nn * 4;
  float*    z    = p;             p += (size_t)Nn * 4;

  float* emb   = (float*)d_out;            // [N,64]
  float* preds = (float*)d_out + (size_t)Nn * 64;

  const unsigned NEGINF_FLIP = 0x007FFFFFu;  // fflip(-inf)

  // ================= Layer 1: 128 -> 4x64 (concat) =================
  {
    int waves = (Nn / 16) * (256 / 64);
    gemm_wmma<128, 256, 4><<<cdiv(waves, 8), 256, 0, stream>>>(x, W1, bufA, Nn);
    attn_logits<256, 4><<<cdiv(Nn, 8), 256, 0, stream>>>(bufA, a1s, a1d, als, ald, Nn);
    init_u32<<<cdiv((long long)Nn * 4, 256), 256, 0, stream>>>(m_u, NEGINF_FLIP, (long long)Nn * 4);
    init_u32<<<cdiv((long long)Nn * 4, 256), 256, 0, stream>>>((unsigned*)z, 0u, (long long)Nn * 4);
    init_u32<<<cdiv((long long)Nn * 256, 256), 256, 0, stream>>>((unsigned*)bufB, 0u, (long long)Nn * 256);
    edge_max<4><<<cdiv((long long)ET * 4, 256), 256, 0, stream>>>(esrc, edst, E, Nn, als, ald, m_u);
    edge_sum<4><<<cdiv((long long)ET * 4, 256), 256, 0, stream>>>(esrc, edst, E, Nn, als, ald, m_u, z);
    edge_scatter<256, 4><<<ET, 256, 0, stream>>>(esrc, edst, E, Nn, als, ald, m_u, z, bufA, bufB);
    bias_relu<<<cdiv((long long)Nn * 256, 256), 256, 0, stream>>>(bufB, b1, 256, (long long)Nn * 256);
  }

  // ================= Layer 2: 256 -> 4x64 (concat) =================
  {
    int waves = (Nn / 16) * (256 / 64);
    gemm_wmma<256, 256, 4><<<cdiv(waves, 8), 256, 0, stream>>>(bufB, W2, bufA, Nn);
    attn_logits<256, 4><<<cdiv(Nn, 8), 256, 0, stream>>>(bufA, a2s, a2d, als, ald, Nn);
    init_u32<<<cdiv((long long)Nn * 4, 256), 256, 0, stream>>>(m_u, NEGINF_FLIP, (long long)Nn * 4);
    init_u32<<<cdiv((long long)Nn * 4, 256), 256, 0, stream>>>((unsigned*)z, 0u, (long long)Nn * 4);
    init_u32<<<cdiv((long long)Nn * 256, 256), 256, 0, stream>>>((unsigned*)bufB, 0u, (long long)Nn * 256);
    edge_max<4><<<cdiv((long long)ET * 4, 256), 256, 0, stream>>>(esrc, edst, E, Nn, als, ald, m_u);
    edge_sum<4><<<cdiv((long long)ET * 4, 256), 256, 0, stream>>>(esrc, edst, E, Nn, als, ald, m_u, z);
    edge_scatter<256, 4><<<ET, 256, 0, stream>>>(esrc, edst, E, Nn, als, ald, m_u, z, bufA, bufB);
    bias_relu<<<cdiv((long long)Nn * 256, 256), 256, 0, stream>>>(bufB, b2, 256, (long long)Nn * 256);
  }

  // ================= Layer 3: 256 -> 64 (1 head, mean==identity) ===
  {
    int waves = (Nn / 16) * (64 / 64);
    gemm_wmma<256, 64, 4><<<cdiv(waves, 8), 256, 0, stream>>>(bufB, W3, bufA, Nn);
    attn_logits<64, 1><<<cdiv(Nn, 8), 256, 0, stream>>>(bufA, a3s, a3d, als, ald, Nn);
    init_u32<<<cdiv((long long)Nn, 256), 256, 0, stream>>>(m_u, NEGINF_FLIP, (long long)Nn);
    init_u32<<<cdiv((long long)Nn, 256), 256, 0, stream>>>((unsigned*)z, 0u, (long long)Nn);
    init_u32<<<cdiv((long long)Nn * 64, 256), 256, 0, stream>>>((unsigned*)emb, 0u, (long long)Nn * 64);
    edge_max<1><<<cdiv((long long)ET, 256), 256, 0, stream>>>(esrc, edst, E, Nn, als, ald, m_u);
    edge_sum<1><<<cdiv((long long)ET, 256), 256, 0, stream>>>(esrc, edst, E, Nn, als, ald, m_u, z);
    edge_scatter<64, 1><<<cdiv(ET, 4), 256, 0, stream>>>(esrc, edst, E, Nn, als, ald, m_u, z, bufA, emb);
    finalize<<<cdiv(Nn, 8), 256, 0, stream>>>(emb, b3, preds, Nn);
  }
}